// PixelGNN_8091718385775
// MI455X (gfx1250) — compile-verified
//
#include <hip/hip_runtime.h>

typedef __attribute__((ext_vector_type(2))) float v2f;
typedef __attribute__((ext_vector_type(8))) float v8f;

#define THREADS 256
#define WAVES_PER_BLOCK (THREADS / 32)
#define EDGES_PER_CHUNK 32   // one wave-load of metadata covers 16 pairs

// ---------------------------------------------------------------------------
// Zero-fill d_out (harness poisons it with 0xAA; atomics need a clean base).
// ---------------------------------------------------------------------------
__global__ void __launch_bounds__(THREADS)
zero_f32(float* __restrict__ p, int n) {
  int i = blockIdx.x * blockDim.x + threadIdx.x;
  int stride = gridDim.x * blockDim.x;
  for (; i < n; i += stride) p[i] = 0.0f;
}

// ---------------------------------------------------------------------------
// Two edges per WMMA problem:
//   A = blockdiag(W[t0], W[t1])  (16x16 over K)
//   B = [x[s0]; x[s1]]           (16x16, rows = FIN pair, cols = batch)
//   C = [bias(t0); bias(t1)]     (bias fused into the accumulate)
// Chained as 4 x v_wmma_f32_16x16x4_f32. D rows 0-7 -> edge0 message
// (lanes 0-15), rows 8-15 -> edge1 (lanes 16-31); scattered with native
// global_atomic_add_f32. Metadata for 32 edges is fetched cooperatively
// (3 coalesced loads) and extracted per-pair with v_readlane.
// FIN=8, FOUT=8, B=16 -> per-node x / y slab stride = 128 floats.
// ---------------------------------------------------------------------------
__global__ void __launch_bounds__(THREADS)
gnn_edge_wmma(const float* __restrict__ x,    // [N, 8, 16]
              const float* __restrict__ Wt,   // [T, 8, 8]
              const float* __restrict__ bt,   // [T, 8]
              const int* __restrict__ src,
              const int* __restrict__ tgt,
              const int* __restrict__ etype,
              float* __restrict__ out,        // [N, 8, 16]
              int E) {
  const int lane = threadIdx.x & 31;
  const int col  = lane & 15;   // matrix column (batch) / A-row M
  const int half = lane >> 4;   // K sub-slice select within each 4-wide step
  const int rb   = 2 * half;    // K base: lanes 0-15 -> {k0,k0+1}, 16-31 -> {k0+2,k0+3}
  const bool lo  = (col < 8);   // A-rows 0-7 (edge0 W) vs 8-15 (edge1 W)

  const int wave   = (int)(blockIdx.x * WAVES_PER_BLOCK + (threadIdx.x >> 5));
  const int nwav   = (int)(gridDim.x * WAVES_PER_BLOCK);
  const int nchunk = (E + EDGES_PER_CHUNK - 1) / EDGES_PER_CHUNK;

  for (int ch = wave; ch < nchunk; ch += nwav) {
    const int base = ch * EDGES_PER_CHUNK;

    // ---- Cooperative metadata fetch: lane L holds edge (base+L) ----
    int eL = base + lane;
    if (eL > E - 1) eL = E - 1;            // clamp keeps EXEC all-ones
    const int sv = src[eL];
    const int gv = tgt[eL];
    const int tv = etype[eL] - 1;

    int npair = (E - base + 1) >> 1;
    if (npair > EDGES_PER_CHUNK / 2) npair = EDGES_PER_CHUNK / 2;

    for (int q = 0; q < npair; ++q) {
      const int l0 = 2 * q, l1 = 2 * q + 1;
      const bool has1 = (base + l1 < E);
      const int s0 = __builtin_amdgcn_readlane(sv, l0);
      const int g0 = __builtin_amdgcn_readlane(gv, l0);
      const int t0 = __builtin_amdgcn_readlane(tv, l0);
      const int s1 = __builtin_amdgcn_readlane(sv, l1);
      const int g1 = __builtin_amdgcn_readlane(gv, l1);
      const int t1 = __builtin_amdgcn_readlane(tv, l1);

      const float* x0 = x + (size_t)s0 * 128;
      const float* x1 = x + (size_t)s1 * 128;

      // ---- B fragments: rows 0-7 from x[s0], rows 8-15 from x[s1] ----
      const float bf0 = x0[(rb + 0) * 16 + col];  // step K=0..3
      const float bf1 = x0[(rb + 1) * 16 + col];
      const float bf2 = x0[(rb + 4) * 16 + col];  // step K=4..7
      const float bf3 = x0[(rb + 5) * 16 + col];
      const float bf4 = x1[(rb + 0) * 16 + col];  // step K=8..11
      const float bf5 = x1[(rb + 1) * 16 + col];
      const float bf6 = x1[(rb + 4) * 16 + col];  // step K=12..15
      const float bf7 = x1[(rb + 5) * 16 + col];

      // ---- A fragments: select the row pointer, load unconditionally,
      //      then mask into the block-diagonal positions (no branches) ----
      const int widx = lo ? (t0 * 64 + col * 8) : (t1 * 64 + (col & 7) * 8);
      const float* wr = Wt + widx;
      const float wv0 = wr[rb + 0];
      const float wv1 = wr[rb + 1];
      const float wv2 = wr[rb + 4];
      const float wv3 = wr[rb + 5];
      const float z = 0.0f;
      v2f a01 = {lo ? wv0 : z, lo ? wv1 : z};   // K slice 0..3   (W[t0] block)
      v2f a23 = {lo ? wv2 : z, lo ? wv3 : z};   // K slice 4..7
      v2f a45 = {lo ? z : wv0, lo ? z : wv1};   // K slice 8..11  (W[t1] block)
      v2f a67 = {lo ? z : wv2, lo ? z : wv3};   // K slice 12..15

      // ---- C init with bias (2 x b128 per lane) ----
      const float* bp = bt + (size_t)(half ? t1 : t0) * 8;
      v8f c;
#pragma unroll
      for (int v = 0; v < 8; ++v) c[v] = bp[v];

      // ---- 16x16x16 FP32 GEMM as 4 chained 16x16x4 WMMAs ----
      v2f b01 = {bf0, bf1}, b23 = {bf2, bf3};
      v2f b45 = {bf4, bf5}, b67 = {bf6, bf7};
      c = __builtin_amdgcn_wmma_f32_16x16x4_f32(false, a01, false, b01, (short)0, c, false, false);
      c = __builtin_amdgcn_wmma_f32_16x16x4_f32(false, a23, false, b23, (short)0, c, false, false);
      c = __builtin_amdgcn_wmma_f32_16x16x4_f32(false, a45, false, b45, (short)0, c, false, false);
      c = __builtin_amdgcn_wmma_f32_16x16x4_f32(false, a67, false, b67, (short)0, c, false, false);

      // ---- Scatter: lanes 0-15 -> edge0, lanes 16-31 -> edge1 ----
      if (half == 0 || has1) {
        float* yp = out + (size_t)(half ? g1 : g0) * 128 + col;
#pragma unroll
        for (int v = 0; v < 8; ++v) {
          unsafeAtomicAdd(yp + v * 16, c[v]);   // native global_atomic_add_f32
        }
      }
    }
  }
}

// ---------------------------------------------------------------------------
extern "C" void kernel_launch(void* const* d_in, const int* in_sizes, int n_in,
                              void* d_out, int out_size, void* d_ws, size_t ws_size,
                              hipStream_t stream) {
  const float* x   = (const float*)d_in[0];  // [N, 8, 16]
  const float* W   = (const float*)d_in[1];  // [T, 8, 8]
  const float* b   = (const float*)d_in[2];  // [T, 8]
  const int*   src = (const int*)d_in[3];    // [E]
  const int*   tgt = (const int*)d_in[4];    // [E]
  const int*   et  = (const int*)d_in[5];    // [E]
  const int    E   = in_sizes[3];
  float* out = (float*)d_out;

  // Zero the accumulator surface (32 MB -> L2-resident).
  int zblocks = (out_size + THREADS * 8 - 1) / (THREADS * 8);
  if (zblocks > 4096) zblocks = 4096;
  if (zblocks < 1) zblocks = 1;
  zero_f32<<<zblocks, THREADS, 0, stream>>>(out, out_size);

  const int nchunk = (E + EDGES_PER_CHUNK - 1) / EDGES_PER_CHUNK;
  int blocks = (nchunk + WAVES_PER_BLOCK - 1) / WAVES_PER_BLOCK;
  if (blocks > 16384) blocks = 16384;  // grid-stride loop covers the rest
  if (blocks < 1) blocks = 1;
  gnn_edge_wmma<<<blocks, THREADS, 0, stream>>>(x, W, b, src, tgt, et, out, E);
}